// PIMARL_actor_26474178412687
// MI455X (gfx1250) — compile-verified
//
#include <hip/hip_runtime.h>
#include <hip/hip_bf16.h>

typedef float v2f __attribute__((ext_vector_type(2)));
typedef float v8f __attribute__((ext_vector_type(8)));

__device__ inline v8f wmma4(v2f a, v2f b, v8f c) {
    // D = A(16x4 f32) * B(4x16 f32) + C   -> v_wmma_f32_16x16x4_f32
    return __builtin_amdgcn_wmma_f32_16x16x4_f32(false, a, false, b, (short)0, c, false, false);
}

// hardware reciprocal (v_rcp_f32) -- avoids the 7-op IEEE div fixup sequence
__device__ inline float frcp_(float x) { return __builtin_amdgcn_rcpf(x); }
__device__ inline float sigm_(float x) { return frcp_(1.0f + __expf(-x)); }
__device__ inline float silu_(float x) { return x * sigm_(x); }

#define NS 8   // samples per block in the H kernel

// ---------------------------------------------------------------------------
// Kernel 1: H module forward-mode JVP. One block = NS flat samples.
// Produces g0[s] = dH_s/dx[s,0], g1[s] = dH_s/dx[s,1]   (s = 0..4095)
// ---------------------------------------------------------------------------
__global__ __launch_bounds__(256) void h_grad_kernel(
    const float* __restrict__ x,
    const float* __restrict__ Win, const float* __restrict__ bin,
    const float* __restrict__ Aq,  const float* __restrict__ Ak, const float* __restrict__ Av,
    const float* __restrict__ Bq,  const float* __restrict__ Bk, const float* __restrict__ Bv,
    const float* __restrict__ Wout, const float* __restrict__ bout,
    float* __restrict__ g0, float* __restrict__ g1)
{
    __shared__ float Xs[NS * 16];
    __shared__ float hB[NS * 128], dh0B[NS * 128], dh1B[NS * 128];
    __shared__ float qB[NS * 128], kB[NS * 128], vB[NS * 128];
    __shared__ float dq0B[NS * 128], dk0B[NS * 128], dv0B[NS * 128];
    __shared__ float dq1B[NS * 128], dk1B[NS * 128], dv1B[NS * 128];
    __shared__ float zy[NS * 25], zd0[NS * 25], zd1[NS * 25];

    const int tid  = threadIdx.x;
    const int lane = tid & 31;
    const int wave = tid >> 5;
    const int s0   = blockIdx.x * NS;   // first flat sample of this block

    // ---- Phase A: load X tile --------------------------------------------
    if (tid < NS * 16) Xs[tid] = x[s0 * 16 + tid];
    __syncthreads();

    // ---- Phase B: S = X @ Win + bin ; h = silu(S); dh_t = silu'(S)*Win[t,:]
    {
        const int d0 = wave * 16;
        const int n  = lane & 15;
        v8f acc = {0.f,0.f,0.f,0.f,0.f,0.f,0.f,0.f};
        for (int k0 = 0; k0 < 16; k0 += 4) {
            const int kk = k0 + ((lane >> 4) << 1);
            v2f a; a.x = Xs[(lane & 7) * 16 + kk]; a.y = Xs[(lane & 7) * 16 + kk + 1];
            v2f b; b.x = Win[kk * 128 + d0 + n];   b.y = Win[(kk + 1) * 128 + d0 + n];
            acc = wmma4(a, b, acc);
        }
        if (lane < 16) {                 // rows 0..7 (valid samples) live on lanes 0..15
            const int d = d0 + n;
            const float w0 = Win[d], w1 = Win[128 + d], bb = bin[d];
#pragma unroll
            for (int r = 0; r < NS; ++r) {
                const float pre = acc[r] + bb;
                const float sg  = sigm_(pre);
                hB[r * 128 + d]   = pre * sg;
                const float ds    = sg * (1.f + pre * (1.f - sg));
                dh0B[r * 128 + d] = ds * w0;
                dh1B[r * 128 + d] = ds * w1;
            }
        }
    }
    __syncthreads();

    // ---- Phase C: nine GEMMs (q,k,v) x (primal,t0,t1), M=8 N=128 K=128 ----
    for (int job = wave; job < 72; job += 8) {
        const int g  = job >> 3;          // gemm id 0..8
        const int nt = job & 7;           // N tile
        const int vr = g / 3;             // 0 primal, 1 tangent0, 2 tangent1
        const int m3 = g % 3;             // 0 q, 1 k, 2 v
        const float* A = (vr == 0) ? hB : (vr == 1) ? dh0B : dh1B;
        const float* W = (m3 == 0) ? Aq : (m3 == 1) ? Ak : Av;
        const float* BB = (m3 == 0) ? Bq : (m3 == 1) ? Bk : Bv;
        float* D;
        switch (g) {
            case 0: D = qB;  break; case 1: D = kB;  break; case 2: D = vB;  break;
            case 3: D = dq0B; break; case 4: D = dk0B; break; case 5: D = dv0B; break;
            case 6: D = dq1B; break; case 7: D = dk1B; break; default: D = dv1B; break;
        }
        const int n0 = nt * 16;
        const int n  = lane & 15;
        v8f acc = {0.f,0.f,0.f,0.f,0.f,0.f,0.f,0.f};
        for (int k0 = 0; k0 < 128; k0 += 4) {
            const int kk = k0 + ((lane >> 4) << 1);
            v2f a; a.x = A[(lane & 7) * 128 + kk];   a.y = A[(lane & 7) * 128 + kk + 1];
            v2f b; b.x = W[(n0 + n) * 128 + kk];     b.y = W[(n0 + n) * 128 + kk + 1]; // B = W^T
            acc = wmma4(a, b, acc);
        }
        if (lane < 16) {
            const float bias = (vr == 0) ? BB[n0 + n] : 0.f;
#pragma unroll
            for (int r = 0; r < NS; ++r)
                D[r * 128 + n0 + n] = acc[r] + bias;   // pre-activation (primal) / raw jvp
        }
    }
    __syncthreads();

    // ---- Phase C3: apply silu and scale tangents by silu'(pre) -----------
    for (int idx = tid; idx < NS * 128; idx += 256) {
        { float p = qB[idx]; float s = sigm_(p); float d = s*(1.f+p*(1.f-s));
          qB[idx] = p*s; dq0B[idx] *= d; dq1B[idx] *= d; }
        { float p = kB[idx]; float s = sigm_(p); float d = s*(1.f+p*(1.f-s));
          kB[idx] = p*s; dk0B[idx] *= d; dk1B[idx] *= d; }
        { float p = vB[idx]; float s = sigm_(p); float d = s*(1.f+p*(1.f-s));
          vB[idx] = p*s; dv0B[idx] *= d; dv1B[idx] *= d; }
    }
    __syncthreads();

    // ---- Phase D: attention rows + softmax JVP (wave = sample, lane = row)
    {
        const int s = wave;
        const float* kp  = &kB[s * 128];   const float* vp  = &vB[s * 128];
        const float* k0p = &dk0B[s * 128]; const float* v0p = &dv0B[s * 128];
        const float* k1p = &dk1B[s * 128]; const float* v1p = &dv1B[s * 128];
        float kmx = -1e30f, kmn = 1e30f;
        for (int j = lane; j < 128; j += 32) { float kv = kp[j]; kmx = fmaxf(kmx, kv); kmn = fminf(kmn, kv); }
        for (int off = 16; off; off >>= 1) {
            kmx = fmaxf(kmx, __shfl_xor(kmx, off, 32));
            kmn = fminf(kmn, __shfl_xor(kmn, off, 32));
        }
        for (int rc = 0; rc < 4; ++rc) {
            const int i = rc * 32 + lane;
            const float qi = qB[s*128+i], dq0i = dq0B[s*128+i], dq1i = dq1B[s*128+i];
            const float mi = fmaxf(qi * kmx, qi * kmn);   // exact row max of qi*kj
            float se=0, sv=0, skv=0, sk=0, a0kv=0, a0k=0, a0v=0, a1kv=0, a1k=0, a1v=0;
#pragma unroll 4
            for (int j = 0; j < 128; ++j) {
                const float kj = kp[j], vj = vp[j];
                const float e = __expf(qi * kj - mi);
                se += e; sv += e * vj; skv += e * kj * vj; sk += e * kj;
                const float x0 = k0p[j], y0 = v0p[j], x1 = k1p[j], y1 = v1p[j];
                a0kv += e * x0 * vj; a0k += e * x0; a0v += e * y0;
                a1kv += e * x1 * vj; a1k += e * x1; a1v += e * y1;
            }
            const float inv = frcp_(se);
            const float u  = sv * inv, A1 = skv * inv, A2 = sk * inv;
            const float du0 = dq0i*(A1 - A2*u) + qi*(a0kv*inv - a0k*inv*u) + a0v*inv;
            const float du1 = dq1i*(A1 - A2*u) + qi*(a1kv*inv - a1k*inv*u) + a1v*inv;
            const float sg = sigm_(u), ds = sg * (1.f + u * (1.f - sg));
            hB[s*128+i]   = u * sg;          // y     (reuse h arrays)
            dh0B[s*128+i] = ds * du0;        // dy/dt0
            dh1B[s*128+i] = ds * du1;        // dy/dt1
        }
    }
    __syncthreads();

    // ---- Phase E: z = Y @ Wout + bout; y2 = silu(z) + tangents ------------
    if (tid < NS * 25) {
        const int s = tid / 25, o = tid % 25;
        float acc = bout[o], a0 = 0.f, a1 = 0.f;
        for (int i = 0; i < 128; ++i) {
            const float w = Wout[i * 25 + o];
            acc += hB[s*128+i] * w; a0 += dh0B[s*128+i] * w; a1 += dh1B[s*128+i] * w;
        }
        const float sg = sigm_(acc), ds = sg * (1.f + acc * (1.f - sg));
        zy[tid] = acc * sg; zd0[tid] = ds * a0; zd1[tid] = ds * a1;
    }
    __syncthreads();

    // ---- Phase F: quadratic head, emit dH/dt0, dH/dt1 ---------------------
    if (tid < NS) {
        const float* Z  = &zy[tid * 25];
        const float* D0 = &zd0[tid * 25];
        const float* D1 = &zd1[tid * 25];
        float m[5], dm0[5], dm1[5];
#pragma unroll
        for (int gI = 0; gI < 5; ++gI) {
            float mm = 0, d0 = 0, d1 = 0;
#pragma unroll
            for (int o = 0; o < 5; ++o) {
                const float yv = Z[gI*5+o];
                mm += yv * yv; d0 += 2.f * yv * D0[gI*5+o]; d1 += 2.f * yv * D1[gI*5+o];
            }
            m[gI] = mm; dm0[gI] = d0; dm1[gI] = d1;
        }
        const float q0=Z[0], q1=Z[1], q2=Z[2], q3=Z[3];
        const float P1 = q0*q0+q1*q1, P2 = q0*q2+q1*q3, P3 = q2*q2+q3*q3;
        {   const float e0=D0[0], e1=D0[1], e2=D0[2], e3=D0[3];
            const float dP1 = 2.f*(q0*e0+q1*e1);
            const float dP2 = e0*q2+q0*e2+e1*q3+q1*e3;
            const float dP3 = 2.f*(q2*e2+q3*e3);
            g0[s0 + tid] = dm0[0]*P1 + m[0]*dP1 + (dm0[1]+dm0[2])*P2 + (m[1]+m[2])*dP2
                         + dm0[3]*P3 + m[3]*dP3 + dm0[4];
        }
        {   const float e0=D1[0], e1=D1[1], e2=D1[2], e3=D1[3];
            const float dP1 = 2.f*(q0*e0+q1*e1);
            const float dP2 = e0*q2+q0*e2+e1*q3+q1*e3;
            const float dP3 = 2.f*(q2*e2+q3*e3);
            g1[s0 + tid] = dm1[0]*P1 + m[0]*dP1 + (dm1[1]+dm1[2])*P2 + (m[1]+m[2])*dP2
                         + dm1[3]*P3 + m[3]*dP3 + dm1[4];
        }
    }
}

// ---------------------------------------------------------------------------
// Kernel 2/3: _att_feat on the 512 UNIQUE samples (jnp.repeat degeneracy).
// OUTD==8  -> writes Jc[b]   = sum of all outputs (J module scalar)
// OUTD==16 -> writes Rf[b*32 + g*8 + j] = sum_{o in group g} out[agent j][o]
// One block = 2 unique samples.
// ---------------------------------------------------------------------------
template<int OUTD>
__global__ __launch_bounds__(256) void attfeat_kernel(
    const float* __restrict__ x,
    const float* __restrict__ Win, const float* __restrict__ bin,
    const float* __restrict__ Aq,  const float* __restrict__ Ak, const float* __restrict__ Av,
    const float* __restrict__ Bq,  const float* __restrict__ Bk, const float* __restrict__ Bv,
    const float* __restrict__ Wout, const float* __restrict__ bout,
    float* __restrict__ outbuf)
{
    __shared__ float Xs[16 * 16];
    __shared__ float hB[16 * 128];              // h rows = sample*8+agent; reused as Y
    __shared__ float qL[128 * 16], kL[128 * 16], vL[128 * 16];
    __shared__ float outv[16 * 16];

    const int tid = threadIdx.x, lane = tid & 31, wave = tid >> 5;
    const int b0  = blockIdx.x * 2;

    // Phase A: X (2 samples x 8 agents x 16 obs = 256 values)
    Xs[tid] = x[b0 * 128 + tid];
    __syncthreads();

    // Phase B: h = silu(X @ Win + bin)   M=16 K=16 N=128
    {
        const int d0 = wave * 16, n = lane & 15;
        v8f acc = {0.f,0.f,0.f,0.f,0.f,0.f,0.f,0.f};
        for (int k0 = 0; k0 < 16; k0 += 4) {
            const int kk = k0 + ((lane >> 4) << 1);
            v2f a; a.x = Xs[(lane & 15) * 16 + kk]; a.y = Xs[(lane & 15) * 16 + kk + 1];
            v2f b; b.x = Win[kk * 128 + d0 + n];    b.y = Win[(kk + 1) * 128 + d0 + n];
            acc = wmma4(a, b, acc);
        }
        const int moff = (lane >> 4) * 8;
        const float bb = bin[d0 + n];
#pragma unroll
        for (int r = 0; r < 8; ++r) {
            const float pre = acc[r] + bb;
            hB[(moff + r) * 128 + d0 + n] = silu_(pre);
        }
    }
    __syncthreads();

    // Phase C: Q/K/V = silu(W @ h' + B)  M=128(i) K=128(j) N=16(2 samples x 8)
    for (int job = wave; job < 24; job += 8) {
        const int m3 = job >> 3, mt = job & 7;
        const float* W  = (m3 == 0) ? Aq : (m3 == 1) ? Ak : Av;
        const float* BB = (m3 == 0) ? Bq : (m3 == 1) ? Bk : Bv;
        float* D = (m3 == 0) ? qL : (m3 == 1) ? kL : vL;
        const int i0 = mt * 16, n = lane & 15;
        v8f acc = {0.f,0.f,0.f,0.f,0.f,0.f,0.f,0.f};
        for (int k0 = 0; k0 < 128; k0 += 4) {
            const int kk = k0 + ((lane >> 4) << 1);
            v2f a; a.x = W[(i0 + (lane & 15)) * 128 + kk]; a.y = W[(i0 + (lane & 15)) * 128 + kk + 1];
            v2f b; b.x = hB[n * 128 + kk];                 b.y = hB[n * 128 + kk + 1];  // B[k][n]=h[n][k]
            acc = wmma4(a, b, acc);
        }
        const int moff = (lane >> 4) * 8;
#pragma unroll
        for (int r = 0; r < 8; ++r) {
            const int i = i0 + moff + r;
            D[i * 16 + n] = silu_(acc[r] + BB[i]);
        }
    }
    __syncthreads();

    // Phase D: per sample attention over j=128 dims, s=8 agents
    {
        const int sb = wave >> 2;
        const int i  = (wave & 3) * 32 + lane;
        const int base = sb * 8;
        float qv[8];
#pragma unroll
        for (int s = 0; s < 8; ++s) qv[s] = qL[i * 16 + base + s];
        float mi = -1e30f;
        for (int j = 0; j < 128; ++j) {
            float t = 0.f;
#pragma unroll
            for (int s = 0; s < 8; ++s) t += qv[s] * kL[j * 16 + base + s];
            mi = fmaxf(mi, t);
        }
        float se = 0.f, acc[8] = {0,0,0,0,0,0,0,0};
        for (int j = 0; j < 128; ++j) {
            float t = 0.f;
#pragma unroll
            for (int s = 0; s < 8; ++s) t += qv[s] * kL[j * 16 + base + s];
            const float e = __expf(t - mi);
            se += e;
#pragma unroll
            for (int s = 0; s < 8; ++s) acc[s] += e * vL[j * 16 + base + s];
        }
        const float inv = frcp_(se);
        __syncthreads();                 // done reading hB in phase C; safe to overwrite
#pragma unroll
        for (int s = 0; s < 8; ++s) {
            const float u = acc[s] * inv;
            hB[(base + s) * 128 + i] = silu_(u);   // Y[row=sample*8+agent][i]
        }
    }
    __syncthreads();

    // Phase E: out = silu(Y @ Wout + bout)  (16 rows x OUTD)
    if (tid < 16 * OUTD) {
        const int row = tid / OUTD, o = tid % OUTD;
        float acc = bout[o];
        for (int i = 0; i < 128; ++i) acc += hB[row * 128 + i] * Wout[i * OUTD + o];
        outv[row * 16 + o] = silu_(acc);
    }
    __syncthreads();

    if constexpr (OUTD == 8) {
        if (tid < 2) {
            float c = 0.f;
            for (int r = 0; r < 8; ++r)
                for (int o = 0; o < 8; ++o) c += outv[(tid * 8 + r) * 16 + o];
            outbuf[b0 + tid] = c;                   // Jc[b]
        }
    } else {
        if (tid < 64) {
            const int sb = tid >> 5, gI = (tid >> 3) & 3, j = tid & 7;
            float a = 0.f;
#pragma unroll
            for (int o = 0; o < 4; ++o) a += outv[(sb * 8 + j) * 16 + gI * 4 + o];
            outbuf[(b0 + sb) * 32 + gI * 8 + j] = a; // Rf[b][group][agent]
        }
    }
}

// ---------------------------------------------------------------------------
// Kernel 4: per batch assemble Rm, u = (J - Rm Rm^T) dHdz.  16 batches/block.
// ---------------------------------------------------------------------------
__global__ __launch_bounds__(256) void assemble_kernel(
    const float* __restrict__ g0v, const float* __restrict__ g1v,
    const float* __restrict__ Jc,  const float* __restrict__ Rf,
    float* __restrict__ out)
{
    __shared__ float RmS[16][16 * 17];
    __shared__ float dS[16][16];
    __shared__ float wS[16][16];
    const int bl = threadIdx.x >> 4;
    const int i  = threadIdx.x & 15;
    const int b  = blockIdx.x * 16 + bl;
    const float* A = &Rf[b * 32];  // a1[0..7] a2[0..7] a3[0..7] a4[0..7]

    const float di = (i < 8) ? g0v[b * 8 + i] : g1v[b * 8 + i - 8];
    dS[bl][i] = di;

    float row[16];
#pragma unroll
    for (int j = 0; j < 16; ++j) {
        const float rij = (i < 8) ? A[j] : A[16 + j];    // R[i][j] (block rows share values)
        const float rji = (j < 8) ? A[i] : A[16 + i];    // R[j][i]
        row[j] = rij * rij + rji * rji;                  // Rupper[i][j]
    }
    float offsum = 0.f;
#pragma unroll
    for (int j = 0; j < 16; ++j) if (j != i) offsum += row[j];
#pragma unroll
    for (int j = 0; j < 16; ++j)
        RmS[bl][i * 17 + j] = (j == i) ? (row[i] + offsum) : (-row[j]);
    __syncthreads();

    float wk = 0.f;
#pragma unroll
    for (int j = 0; j < 16; ++j) wk += RmS[bl][j * 17 + i] * dS[bl][j];
    wS[bl][i] = wk;
    __syncthreads();

    const float c = Jc[b];
    float u = (i < 8) ? 2.f * c * dS[bl][8 + i] : -2.f * c * dS[bl][i - 8];
#pragma unroll
    for (int k = 0; k < 16; ++k) u -= RmS[bl][i * 17 + k] * wS[bl][k];
    out[b * 16 + i] = u;
}

// ---------------------------------------------------------------------------
extern "C" void kernel_launch(void* const* d_in, const int* in_sizes, int n_in,
                              void* d_out, int out_size, void* d_ws, size_t ws_size,
                              hipStream_t stream) {
    const float* x = (const float*)d_in[0];
    const float* H_Win = (const float*)d_in[1];  const float* H_bin = (const float*)d_in[2];
    const float* H_Aq  = (const float*)d_in[3];  const float* H_Ak  = (const float*)d_in[4];
    const float* H_Av  = (const float*)d_in[5];  const float* H_Bq  = (const float*)d_in[6];
    const float* H_Bk  = (const float*)d_in[7];  const float* H_Bv  = (const float*)d_in[8];
    const float* H_Wout = (const float*)d_in[9]; const float* H_bout = (const float*)d_in[10];
    const float* J_Win = (const float*)d_in[11]; const float* J_bin = (const float*)d_in[12];
    const float* J_Aq  = (const float*)d_in[13]; const float* J_Ak  = (const float*)d_in[14];
    const float* J_Av  = (const float*)d_in[15]; const float* J_Bq  = (const float*)d_in[16];
    const float* J_Bk  = (const float*)d_in[17]; const float* J_Bv  = (const float*)d_in[18];
    const float* J_Wout = (const float*)d_in[19]; const float* J_bout = (const float*)d_in[20];
    const float* R_Win = (const float*)d_in[21]; const float* R_bin = (const float*)d_in[22];
    const float* R_Aq  = (const float*)d_in[23]; const float* R_Ak  = (const float*)d_in[24];
    const float* R_Av  = (const float*)d_in[25]; const float* R_Bq  = (const float*)d_in[26];
    const float* R_Bk  = (const float*)d_in[27]; const float* R_Bv  = (const float*)d_in[28];
    const float* R_Wout = (const float*)d_in[29]; const float* R_bout = (const float*)d_in[30];

    float* ws = (float*)d_ws;
    float* g0 = ws;               // 4096
    float* g1 = ws + 4096;        // 4096
    float* Jc = ws + 8192;        // 512
    float* Rf = ws + 8704;        // 512*32

    h_grad_kernel<<<512, 256, 0, stream>>>(x, H_Win, H_bin, H_Aq, H_Ak, H_Av,
                                           H_Bq, H_Bk, H_Bv, H_Wout, H_bout, g0, g1);
    attfeat_kernel<8><<<256, 256, 0, stream>>>(x, J_Win, J_bin, J_Aq, J_Ak, J_Av,
                                               J_Bq, J_Bk, J_Bv, J_Wout, J_bout, Jc);
    attfeat_kernel<16><<<256, 256, 0, stream>>>(x, R_Win, R_bin, R_Aq, R_Ak, R_Av,
                                                R_Bq, R_Bk, R_Bv, R_Wout, R_bout, Rf);
    assemble_kernel<<<32, 256, 0, stream>>>(g0, g1, Jc, Rf, (float*)d_out);
}